// Head_27290222198876
// MI455X (gfx1250) — compile-verified
//
#include <hip/hip_runtime.h>

// Problem constants (from setup_inputs): B=4, T=4096, C=1024, H=64
#define B_DIM  4
#define T_DIM  4096
#define C_DIM  1024
#define H_DIM  64
#define BT     (B_DIM * T_DIM)          // 16384 rows
#define NCHUNK 64                        // T-chunks per batch for k^T v
#define CHUNK  (T_DIM / NCHUNK)          // 64 timesteps per chunk
#define WPACK  (C_DIM * H_DIM)           // packed floats per weight matrix (65536)
#define MPACK  (H_DIM * H_DIM)           // packed floats per batch M (4096)

typedef __attribute__((ext_vector_type(2))) float v2f;
typedef __attribute__((ext_vector_type(4))) float v4f;
typedef __attribute__((ext_vector_type(8))) float v8f;

static __device__ __forceinline__ v8f wmma_f32(v2f a, v2f b, v8f c) {
    // D = A(16x4) * B(4x16) + C(16x16), fp32 WMMA on CDNA5
    return __builtin_amdgcn_wmma_f32_16x16x4_f32(
        /*neg_a=*/false, a, /*neg_b=*/false, b,
        /*c_mod=*/(short)0, c, /*reuse_a=*/false, /*reuse_b=*/false);
}

// Packed B-fragment layout for a KxN (K-major, row stride N=64) operand:
//   packed[((s*2 + g)*32 + lane)*4 + c]
//     s    = K-step (4 K values per step)
//     g    = column group: g0 -> ntiles 0,1 ; g1 -> ntiles 2,3
//     lane = ph*16 + li  (ph = K-pair select, li = column-in-tile)
//     c    = {nt_lo k0, nt_lo k1, nt_hi k0, nt_hi k1}
// i.e. element (s,g,l,c) = SRC[4s + 2*ph + (c&1)][g*32 + (c>>1)*16 + li]
// One b128 per lane per (s,g) delivers two whole 4x16 B fragments, coalesced.

// ---------------------------------------------------------------------------
// Kernel 0: repack Wq|Wk|Wv [1024x64] into fragment-native layout (768 KB)
// ---------------------------------------------------------------------------
__global__ void repack_w_kernel(const float* __restrict__ Wq,
                                const float* __restrict__ Wk,
                                const float* __restrict__ Wv,
                                float* __restrict__ Wp)
{
    const int e = blockIdx.x * blockDim.x + threadIdx.x;
    if (e >= 3 * WPACK) return;
    const int m = e / WPACK;
    const int o = e % WPACK;
    const int c  = o & 3;
    const int l  = (o >> 2) & 31;
    const int g  = (o >> 7) & 1;
    const int s  = o >> 8;
    const int ph = l >> 4, li = l & 15;
    const int row = 4 * s + 2 * ph + (c & 1);
    const int col = g * 32 + ((c >> 1) & 1) * 16 + li;
    const float* W = (m == 0) ? Wq : (m == 1) ? Wk : Wv;
    Wp[e] = W[row * H_DIM + col];
}

// ---------------------------------------------------------------------------
// Kernel 1: fused projections  y = x @ W + b  for W in {Wq,Wk,Wv}
// grid.x = BT/16 stripes, block = 96 threads = 3 waves (wave w -> matrix w)
// Inner step: 1x b64 (x) + 2x b128 (packed W) + 4x WMMA.
// ---------------------------------------------------------------------------
__global__ void proj_kernel(const float* __restrict__ x,
                            const float* __restrict__ Wp,
                            const float* __restrict__ bq,
                            const float* __restrict__ bk,
                            const float* __restrict__ bv,
                            float* __restrict__ q, float* __restrict__ k,
                            float* __restrict__ v)
{
    const int lane = threadIdx.x & 31;
    const int wid  = threadIdx.x >> 5;      // 0:q 1:k 2:v
    const int ph   = lane >> 4;             // half-wave -> K pair select
    const int li   = lane & 15;             // row (A) / col (B,D) index
    const int row0 = blockIdx.x * 16;

    const float* bias;  float* y;
    if (wid == 0)      { bias = bq; y = q; }
    else if (wid == 1) { bias = bk; y = k; }
    else               { bias = bv; y = v; }

    v8f acc0 = {}, acc1 = {}, acc2 = {}, acc3 = {};

    // A frag: x[row0+li][4s + 2*ph + {0,1}]
    const float* xrow = x  + (size_t)(row0 + li) * C_DIM + 2 * ph;
    const float* wp   = Wp + (size_t)wid * WPACK + lane * 4;

#pragma unroll 4
    for (int s = 0; s < C_DIM / 4; ++s) {
        v2f a  = *(const v2f*)(xrow + 4 * s);
        v4f g0 = *(const v4f*)(wp + s * 256);
        v4f g1 = *(const v4f*)(wp + s * 256 + 128);
        v2f b0 = {g0.x, g0.y}, b1 = {g0.z, g0.w};
        v2f b2 = {g1.x, g1.y}, b3 = {g1.z, g1.w};
        acc0 = wmma_f32(a, b0, acc0);
        acc1 = wmma_f32(a, b1, acc1);
        acc2 = wmma_f32(a, b2, acc2);
        acc3 = wmma_f32(a, b3, acc3);
    }

    const float bb0 = bias[li + 0];
    const float bb1 = bias[li + 16];
    const float bb2 = bias[li + 32];
    const float bb3 = bias[li + 48];

#pragma unroll
    for (int r = 0; r < 8; ++r) {
        const int row = row0 + r + 8 * ph;         // D row layout
        float* yp = y + (size_t)row * H_DIM + li;
        yp[0]  = acc0[r] + bb0;
        yp[16] = acc1[r] + bb1;
        yp[32] = acc2[r] + bb2;
        yp[48] = acc3[r] + bb3;
    }
}

// ---------------------------------------------------------------------------
// Kernel 2: partial M = k^T v per (batch, T-chunk).  One wave per block.
// Writes 64x64 fp32 partial to Mpart[blockIdx.x] (deterministic, no atomics).
// ---------------------------------------------------------------------------
__global__ void kvouter_kernel(const float* __restrict__ k,
                               const float* __restrict__ v,
                               float* __restrict__ Mpart)
{
    const int lane = threadIdx.x & 31;
    const int ph   = lane >> 4;
    const int li   = lane & 15;
    const int b    = blockIdx.x / NCHUNK;
    const int c    = blockIdx.x % NCHUNK;
    const int t0   = c * CHUNK;

    const float* kb = k + (size_t)b * T_DIM * H_DIM;
    const float* vb = v + (size_t)b * T_DIM * H_DIM;

    v8f acc[4][4];
#pragma unroll
    for (int i = 0; i < 4; ++i)
#pragma unroll
        for (int j = 0; j < 4; ++j)
            acc[i][j] = (v8f){};

#pragma unroll 2
    for (int tt = 0; tt < CHUNK; tt += 4) {
        const int t = t0 + tt + 2 * ph;
        const float* kp = kb + (size_t)t * H_DIM;   // rows t, t+1
        const float* vp = vb + (size_t)t * H_DIM;
        v2f af[4], bf[4];
#pragma unroll
        for (int i = 0; i < 4; ++i) {
            // A = K^T tile: A[h1][td] = k[t][h1]
            af[i].x = kp[i * 16 + li];
            af[i].y = kp[H_DIM + i * 16 + li];
            // B = V tile:  B[td][h2] = v[t][h2]
            bf[i].x = vp[i * 16 + li];
            bf[i].y = vp[H_DIM + i * 16 + li];
        }
#pragma unroll
        for (int i = 0; i < 4; ++i)
#pragma unroll
            for (int j = 0; j < 4; ++j)
                acc[i][j] = wmma_f32(af[i], bf[j], acc[i][j]);
    }

    float* Mp = Mpart + (size_t)blockIdx.x * MPACK;
#pragma unroll
    for (int i = 0; i < 4; ++i)
#pragma unroll
        for (int j = 0; j < 4; ++j)
#pragma unroll
            for (int r = 0; r < 8; ++r) {
                const int h1 = i * 16 + r + 8 * ph;
                const int h2 = j * 16 + li;
                Mp[h1 * H_DIM + h2] = acc[i][j][r];
            }
}

// ---------------------------------------------------------------------------
// Kernel 3: deterministic reduction of NCHUNK partials, emitted directly in
// the packed B-fragment layout consumed by out_kernel.
// ---------------------------------------------------------------------------
__global__ void reduce_kernel(const float* __restrict__ Mpart,
                              float* __restrict__ Mp)
{
    const int e = blockIdx.x * blockDim.x + threadIdx.x;   // 0 .. B*4096-1
    if (e >= B_DIM * MPACK) return;
    const int b = e / MPACK;
    const int o = e % MPACK;
    // decode packed offset -> (h1, h2)
    const int c  = o & 3;
    const int l  = (o >> 2) & 31;
    const int g  = (o >> 7) & 1;
    const int s  = o >> 8;                  // 0..15
    const int ph = l >> 4, li = l & 15;
    const int h1 = 4 * s + 2 * ph + (c & 1);
    const int h2 = g * 32 + ((c >> 1) & 1) * 16 + li;

    float sum = 0.f;
    const float* p = Mpart + (size_t)b * NCHUNK * MPACK + h1 * H_DIM + h2;
    for (int cc = 0; cc < NCHUNK; ++cc)
        sum += p[(size_t)cc * MPACK];
    Mp[e] = sum;
}

// ---------------------------------------------------------------------------
// Kernel 4: out = (q @ M) * H^-0.5 .  One wave per 16-row stripe.
// Inner step: 1x b64 (q) + 2x b128 (packed M) + 4x WMMA.
// ---------------------------------------------------------------------------
__global__ void out_kernel(const float* __restrict__ q,
                           const float* __restrict__ Mp,
                           float* __restrict__ out)
{
    const int lane = threadIdx.x & 31;
    const int ph   = lane >> 4;
    const int li   = lane & 15;
    const int row0 = blockIdx.x * 16;
    const int b    = row0 >> 12;            // / T_DIM

    v8f acc0 = {}, acc1 = {}, acc2 = {}, acc3 = {};

    const float* qrow = q  + (size_t)(row0 + li) * H_DIM + 2 * ph;
    const float* mp   = Mp + (size_t)b * MPACK + lane * 4;

#pragma unroll
    for (int s = 0; s < H_DIM / 4; ++s) {
        v2f a  = *(const v2f*)(qrow + 4 * s);
        v4f g0 = *(const v4f*)(mp + s * 256);
        v4f g1 = *(const v4f*)(mp + s * 256 + 128);
        v2f b0 = {g0.x, g0.y}, b1 = {g0.z, g0.w};
        v2f b2 = {g1.x, g1.y}, b3 = {g1.z, g1.w};
        acc0 = wmma_f32(a, b0, acc0);
        acc1 = wmma_f32(a, b1, acc1);
        acc2 = wmma_f32(a, b2, acc2);
        acc3 = wmma_f32(a, b3, acc3);
    }

    const float sc = 0.125f;   // 64^-0.5
#pragma unroll
    for (int r = 0; r < 8; ++r) {
        const int row = row0 + r + 8 * ph;
        float* op = out + (size_t)row * H_DIM + li;
        op[0]  = acc0[r] * sc;
        op[16] = acc1[r] * sc;
        op[32] = acc2[r] * sc;
        op[48] = acc3[r] * sc;
    }
}

// ---------------------------------------------------------------------------
extern "C" void kernel_launch(void* const* d_in, const int* in_sizes, int n_in,
                              void* d_out, int out_size, void* d_ws, size_t ws_size,
                              hipStream_t stream) {
    const float* x  = (const float*)d_in[0];
    const float* Wq = (const float*)d_in[1];
    const float* bq = (const float*)d_in[2];
    const float* Wk = (const float*)d_in[3];
    const float* bk = (const float*)d_in[4];
    const float* Wv = (const float*)d_in[5];
    const float* bv = (const float*)d_in[6];
    float* out = (float*)d_out;

    // Workspace layout (floats):
    //   q, k, v : BT*H each (4 MB each)
    //   Mpart   : B*NCHUNK*H*H (4 MB)
    //   Mp      : B*H*H packed (64 KB)
    //   Wp      : 3*C*H packed (768 KB)
    float* q     = (float*)d_ws;
    float* k     = q + (size_t)BT * H_DIM;
    float* v     = k + (size_t)BT * H_DIM;
    float* Mpart = v + (size_t)BT * H_DIM;
    float* Mp    = Mpart + (size_t)B_DIM * NCHUNK * MPACK;
    float* Wp    = Mp + (size_t)B_DIM * MPACK;

    // 0) repack weights into fragment-native layout (L2-resident, 768 KB)
    repack_w_kernel<<<(3 * WPACK + 255) / 256, 256, 0, stream>>>(Wq, Wk, Wv, Wp);

    // 1) q,k,v projections: 1024 stripes x 3 waves
    proj_kernel<<<BT / 16, 96, 0, stream>>>(x, Wp, bq, bk, bv, q, k, v);

    // 2) per-chunk k^T v partials: 4 batches x 64 chunks, 1 wave each
    kvouter_kernel<<<B_DIM * NCHUNK, 32, 0, stream>>>(k, v, Mpart);

    // 3) deterministic chunk reduction -> packed M
    reduce_kernel<<<(B_DIM * MPACK + 255) / 256, 256, 0, stream>>>(Mpart, Mp);

    // 4) out = q @ M * H^-0.5
    out_kernel<<<BT / 16, 32, 0, stream>>>(q, Mp, out);
}